// Attribute_Decoder_39032662786656
// MI455X (gfx1250) — compile-verified
//
#include <hip/hip_runtime.h>

typedef __attribute__((ext_vector_type(2))) float v2f;
typedef __attribute__((ext_vector_type(8))) float v8f;

#define HID 64  // inner feature width (both layers' GEMM K dim and layer-1 width)

// ---------------- utility kernels ----------------

__global__ void zero_f32_kernel(float* __restrict__ p, long n) {
  long i = (long)blockIdx.x * blockDim.x + threadIdx.x;
  if (i < n) p[i] = 0.0f;
}

__global__ void degree_kernel(const int* __restrict__ src, const int* __restrict__ dst,
                              float* __restrict__ deg_out, float* __restrict__ deg_in,
                              int n_edges) {
  int e = blockIdx.x * blockDim.x + threadIdx.x;
  if (e < n_edges) {
    atomicAdd(&deg_out[src[e]], 1.0f);
    atomicAdd(&deg_in[dst[e]], 1.0f);
  }
}

__global__ void rsqrt_kernel(float* __restrict__ p, long n) {
  long i = (long)blockIdx.x * blockDim.x + threadIdx.x;
  if (i < n) p[i] = rsqrtf(fmaxf(p[i], 1.0f));
}

// dst[i] = src[i] * norm[i / 64]; works in-place (dst == src)
__global__ void scale_rows_kernel(float* __restrict__ dstp, const float* __restrict__ srcp,
                                  const float* __restrict__ norm, long n) {
  long i = (long)blockIdx.x * blockDim.x + threadIdx.x;
  if (i < n) dstp[i] = srcp[i] * norm[i >> 6];
}

// agg[dst[e]] += m[src[e]]  (64 feats per edge; 16 lanes/edge, float4 each)
__global__ void scatter_kernel(const int* __restrict__ src, const int* __restrict__ dst,
                               const float* __restrict__ m, float* __restrict__ agg,
                               int n_edges) {
  long t = (long)blockIdx.x * blockDim.x + threadIdx.x;
  int e = (int)(t >> 4);
  int j = (int)(t & 15) << 2;
  if (e < n_edges) {
    int s = src[e], d = dst[e];
    float4 v = *(const float4*)(m + (long)s * HID + j);
    float* a = agg + (long)d * HID + j;
    atomicAdd(a + 0, v.x);
    atomicAdd(a + 1, v.y);
    atomicAdd(a + 2, v.z);
    atomicAdd(a + 3, v.w);
  }
}

// ---------------- fused norm_in-scale + GEMM (f32 WMMA) + bias + ReLU ----------------
// out[m0:m0+16, :] = relu( (agg[m0:m0+16,:] * norm_in[rows, None]) @ W + b )
// One block per 16-row tile; one wave per 16-column tile (ncols/16 waves).
// A-tile staged in LDS (coalesced), read per the 16x4 f32 A-operand layout:
//   lanes 0-15: K = 4*kb + {0,1}; lanes 16-31: K = 4*kb + {2,3}
// B-operand per ISA: VGPR0 holds K=2*half, VGPR1 holds K=2*half+1, col = lane%16.
__global__ void __launch_bounds__(256) gemm_wmma_relu_kernel(
    const float* __restrict__ agg, const float* __restrict__ norm_in,
    const float* __restrict__ W, const float* __restrict__ bias,
    float* __restrict__ out, int n_rows, int ncols) {
  __shared__ float ldsA[16 * HID];

  const int m0 = blockIdx.x * 16;
  const int tid = threadIdx.x;

  // Stage A tile (pre-scaled by norm_in) into LDS, coalesced over the block.
  for (int i = tid; i < 16 * HID; i += blockDim.x) {
    int r = i >> 6;        // row within tile
    int c = i & (HID - 1); // feature index
    int row = m0 + r;
    float v = 0.0f;
    if (row < n_rows) v = agg[(long)row * HID + c] * norm_in[row];
    ldsA[i] = v;
  }
  __syncthreads();

  const int wave = tid >> 5;
  const int lane = tid & 31;
  const int l = lane & 15;      // column within N-tile / row within M-tile
  const int half = lane >> 4;   // selects K pair (A/B) and M upper half (C/D)
  const int n0 = wave * 16;
  if (n0 >= ncols) return;      // uniform per wave -> EXEC stays all-ones
  const int col = n0 + l;

  v8f c = {};
  #pragma unroll
  for (int kb = 0; kb < HID / 4; ++kb) {
    const int k0 = 4 * kb + 2 * half;
    // A operand: 2 consecutive K values of this lane's row (8B aligned in LDS)
    v2f a = *(const v2f*)(&ldsA[l * HID + k0]);
    // B operand: W[k0][col], W[k0+1][col]
    v2f b;
    b.x = W[(long)k0 * ncols + col];
    b.y = W[(long)(k0 + 1) * ncols + col];
    c = __builtin_amdgcn_wmma_f32_16x16x4_f32(
        /*neg_a=*/false, a, /*neg_b=*/false, b,
        /*c_mod=*/(short)0, c, /*reuse_a=*/false, /*reuse_b=*/false);
  }

  const float bv = bias[col];
  #pragma unroll
  for (int i = 0; i < 8; ++i) {
    int row = m0 + i + 8 * half;   // C/D layout: VGPR i -> M = i + 8*half
    if (row < n_rows) {
      float v = c[i] + bv;
      v = fmaxf(v, 0.0f);
      out[(long)row * ncols + col] = v;
    }
  }
}

// ---------------- host side ----------------

extern "C" void kernel_launch(void* const* d_in, const int* in_sizes, int n_in,
                              void* d_out, int out_size, void* d_ws, size_t ws_size,
                              hipStream_t stream) {
  const float* h  = (const float*)d_in[0];
  const int* src  = (const int*)d_in[1];
  const int* dst  = (const int*)d_in[2];
  const float* W1 = (const float*)d_in[3];
  const float* b1 = (const float*)d_in[4];
  const float* W2 = (const float*)d_in[5];
  const float* b2 = (const float*)d_in[6];
  float* out = (float*)d_out;

  const long n_nodes = (long)in_sizes[0] / HID;
  const int n_edges  = in_sizes[1];
  const int nhid     = in_sizes[4];              // 64  (b1 length)
  const int nfeat    = in_sizes[6];              // 128 (b2 length)

  // Workspace layout (floats): norm_out[N] | norm_in[N] | m[N*64] | agg[N*64]
  float* ws       = (float*)d_ws;
  float* norm_out = ws;
  float* norm_in  = norm_out + n_nodes;
  float* buf_m    = norm_in + n_nodes;
  float* buf_agg  = buf_m + n_nodes * HID;

  const long nz = 2 * n_nodes;          // both norm arrays, contiguous
  const long na = n_nodes * HID;        // feature buffer size
  const long nt = (long)n_edges * 16;   // scatter threads
  const int  mtiles = (int)((n_nodes + 15) / 16);

  // --- shared preprocessing: degrees -> rsqrt norms ---
  zero_f32_kernel<<<(int)((nz + 255) / 256), 256, 0, stream>>>(norm_out, nz);
  degree_kernel<<<(n_edges + 255) / 256, 256, 0, stream>>>(src, dst, norm_out, norm_in, n_edges);
  rsqrt_kernel<<<(int)((nz + 255) / 256), 256, 0, stream>>>(norm_out, nz);

  // --- layer 1 ---
  zero_f32_kernel<<<(int)((na + 255) / 256), 256, 0, stream>>>(buf_agg, na);
  scale_rows_kernel<<<(int)((na + 255) / 256), 256, 0, stream>>>(buf_m, h, norm_out, na);
  scatter_kernel<<<(int)((nt + 255) / 256), 256, 0, stream>>>(src, dst, buf_m, buf_agg, n_edges);
  // x = relu(agg*norm_in @ W1 + b1), written back into buf_m (m no longer needed)
  gemm_wmma_relu_kernel<<<mtiles, 32 * (nhid / 16), 0, stream>>>(
      buf_agg, norm_in, W1, b1, buf_m, (int)n_nodes, nhid);

  // --- layer 2 ---
  zero_f32_kernel<<<(int)((na + 255) / 256), 256, 0, stream>>>(buf_agg, na);
  scale_rows_kernel<<<(int)((na + 255) / 256), 256, 0, stream>>>(buf_m, buf_m, norm_out, na);
  scatter_kernel<<<(int)((nt + 255) / 256), 256, 0, stream>>>(src, dst, buf_m, buf_agg, n_edges);
  gemm_wmma_relu_kernel<<<mtiles, 32 * (nfeat / 16), 0, stream>>>(
      buf_agg, norm_in, W2, b2, out, (int)n_nodes, nfeat);
}